// AGNN_72344429134043
// MI455X (gfx1250) — compile-verified
//
#include <hip/hip_runtime.h>

// AGNN layer for MI455X (gfx1250, wave32).
// Per-edge cosine logits are computed 16-edges-at-a-time as the diagonal of
// V_WMMA_F32_16X16X4_F32 (f32 in / f32 acc -> bit-compatible with reference),
// chained over 8 K-steps to cover D=32. Everything else is L2-resident
// gather/scatter with hardware f32 atomics.

typedef __attribute__((ext_vector_type(2))) float v2f;
typedef __attribute__((ext_vector_type(8))) float v8f;

#define DF 32

// Monotonic float <-> uint mapping so atomicMax(uint) implements float max.
static __device__ __forceinline__ unsigned enc_ord(float f) {
  unsigned u = __float_as_uint(f);
  return (u & 0x80000000u) ? ~u : (u | 0x80000000u);
}
static __device__ __forceinline__ float dec_ord(unsigned e) {
  unsigned u = (e & 0x80000000u) ? (e & 0x7fffffffu) : ~e;
  return __uint_as_float(u);
}

// ---------------- Kernel 0: init out / segsum / segmax ----------------
__global__ void agnn_init(float* __restrict__ out, float* __restrict__ segsum,
                          unsigned* __restrict__ segmax, int n_nodes) {
  int stride = gridDim.x * blockDim.x;
  int i0 = blockIdx.x * blockDim.x + threadIdx.x;
  int total = n_nodes * DF;
  for (int i = i0; i < total; i += stride) out[i] = 0.0f;
  for (int i = i0; i < n_nodes; i += stride) {
    segsum[i] = 0.0f;
    segmax[i] = 0x007FFFFFu;  // enc_ord(-inf)
  }
}

// ---------------- Kernel 1: L2-normalize node features ----------------
// One wave per node; lane = feature.
__global__ void agnn_normalize(const float* __restrict__ x,
                               float* __restrict__ xn, int n_nodes) {
  int wid = (blockIdx.x * blockDim.x + threadIdx.x) >> 5;
  int lane = threadIdx.x & 31;
  if (wid >= n_nodes) return;  // uniform per wave
  float v = x[(long)wid * DF + lane];
  float s = v * v;
#pragma unroll
  for (int off = 16; off > 0; off >>= 1) s += __shfl_xor(s, off, 32);
  float inv = 1.0f / fmaxf(sqrtf(s), 1e-12f);
  xn[(long)wid * DF + lane] = v * inv;
}

// ---------------- Kernel 2: per-edge logits via WMMA diagonal ----------------
// One wave handles 16 edges. A(16x4) = xn[dst] rows, B(4x16) = xn[src] cols.
// A layout (f32 16x4): lanes 0-15 M=0..15 {V0=K0,V1=K1}; lanes 16-31 {V0=K2,V1=K3}.
// B mirrors it -> each lane loads float2 at offset k0+2*half from both rows.
// Diagonal of D(16x16): lanes 0-7 hold n=lane in c[lane];
//                        lanes 24-31 hold n=lane-16 in c[lane-24].
__global__ void agnn_logits(const float* __restrict__ xn,
                            const int* __restrict__ src,
                            const int* __restrict__ dst,
                            const float* __restrict__ beta,
                            float* __restrict__ alpha,
                            unsigned* __restrict__ segmax, int n_edges) {
  int wid = (blockIdx.x * blockDim.x + threadIdx.x) >> 5;
  int lane = threadIdx.x & 31;
  long ebase = (long)wid * 16;
  if (ebase >= n_edges) return;  // uniform per wave; EXEC stays all-1s for WMMA

  int n = lane & 15;
  int half = lane >> 4;
  long e = ebase + n;
  bool valid = (e < n_edges);
  if (!valid) e = n_edges - 1;  // clamp loads; stores are guarded below

  int s = src[e];
  int d = dst[e];
  const float* rowd = xn + (long)d * DF;  // A: dst rows
  const float* rows = xn + (long)s * DF;  // B: src cols

  v8f c = {0.f, 0.f, 0.f, 0.f, 0.f, 0.f, 0.f, 0.f};
#pragma unroll
  for (int k0 = 0; k0 < DF; k0 += 4) {
    v2f a = *(const v2f*)(rowd + k0 + 2 * half);
    v2f b = *(const v2f*)(rows + k0 + 2 * half);
    c = __builtin_amdgcn_wmma_f32_16x16x4_f32(
        /*neg_a=*/false, a, /*neg_b=*/false, b,
        /*c_mod=*/(short)0, c, /*reuse_a=*/false, /*reuse_b=*/false);
  }

  // Pull this lane's diagonal element (select chain -> v_cndmask, no branches).
  int sel = (lane < 16) ? lane : (lane - 24);
  float dot = c[0];
  dot = (sel == 1) ? c[1] : dot;
  dot = (sel == 2) ? c[2] : dot;
  dot = (sel == 3) ? c[3] : dot;
  dot = (sel == 4) ? c[4] : dot;
  dot = (sel == 5) ? c[5] : dot;
  dot = (sel == 6) ? c[6] : dot;
  dot = (sel == 7) ? c[7] : dot;

  float av = beta[0] * dot;
  bool holds_diag = (lane < 8) || (lane >= 24);
  if (holds_diag && valid) {
    alpha[e] = av;  // this lane's edge is exactly ebase + n
    atomicMax(&segmax[d], enc_ord(av));
  }
}

// ---------------- Kernel 3: exp(alpha - max) and segment sum ----------------
__global__ void agnn_expsum(const int* __restrict__ dst,
                            const unsigned* __restrict__ segmax,
                            float* __restrict__ alpha,
                            float* __restrict__ segsum, int n_edges) {
  int e = blockIdx.x * blockDim.x + threadIdx.x;
  if (e >= n_edges) return;
  int d = dst[e];
  float m = dec_ord(segmax[d]);
  float w = __expf(alpha[e] - m);
  alpha[e] = w;
  atomicAdd(&segsum[d], w);
}

// ---------------- Kernel 4: weighted scatter-aggregate ----------------
// One wave per edge; lane = feature. out/x/segsum all L2-resident.
__global__ void agnn_aggregate(const float* __restrict__ x,
                               const int* __restrict__ src,
                               const int* __restrict__ dst,
                               const float* __restrict__ alpha,
                               const float* __restrict__ segsum,
                               float* __restrict__ out, int n_edges) {
  int wid = (blockIdx.x * blockDim.x + threadIdx.x) >> 5;
  int lane = threadIdx.x & 31;
  if (wid >= n_edges) return;
  int s = src[wid];
  int d = dst[wid];
  float coef = alpha[wid] / segsum[d];
  float v = x[(long)s * DF + lane];
  atomicAdd(&out[(long)d * DF + lane], coef * v);
}

static inline size_t align256(size_t v) { return (v + 255) & ~(size_t)255; }

extern "C" void kernel_launch(void* const* d_in, const int* in_sizes, int n_in,
                              void* d_out, int out_size, void* d_ws, size_t ws_size,
                              hipStream_t stream) {
  const float* x    = (const float*)d_in[0];
  const float* beta = (const float*)d_in[1];
  const int*   ei   = (const int*)d_in[2];

  const int n_nodes = in_sizes[0] / DF;      // 100000
  const int n_edges = in_sizes[2] / 2;       // 2000000
  const int* src = ei;                        // edge_index[0]
  const int* dst = ei + n_edges;              // edge_index[1]
  float* out = (float*)d_out;

  // Workspace carve-out (≈21.6 MB): xn | alpha | segmax | segsum
  char* ws = (char*)d_ws;
  size_t off = 0;
  float*    xn     = (float*)(ws + off);    off += align256((size_t)n_nodes * DF * sizeof(float));
  float*    alpha  = (float*)(ws + off);    off += align256((size_t)n_edges * sizeof(float));
  unsigned* segmax = (unsigned*)(ws + off); off += align256((size_t)n_nodes * sizeof(unsigned));
  float*    segsum = (float*)(ws + off);
  (void)ws_size;

  const int BLK = 256;  // 8 waves per block

  // K0: init
  agnn_init<<<4096, BLK, 0, stream>>>(out, segsum, segmax, n_nodes);

  // K1: normalize (one wave per node)
  {
    int waves = n_nodes;
    int blocks = (waves * 32 + BLK - 1) / BLK;
    agnn_normalize<<<blocks, BLK, 0, stream>>>(x, xn, n_nodes);
  }

  // K2: WMMA logits + segment max (one wave per 16 edges)
  {
    int waves = (n_edges + 15) / 16;
    int blocks = (waves * 32 + BLK - 1) / BLK;
    agnn_logits<<<blocks, BLK, 0, stream>>>(xn, src, dst, beta, alpha, segmax, n_edges);
  }

  // K3: exp + segment sum (one thread per edge)
  {
    int blocks = (n_edges + BLK - 1) / BLK;
    agnn_expsum<<<blocks, BLK, 0, stream>>>(dst, segmax, alpha, segsum, n_edges);
  }

  // K4: aggregate (one wave per edge)
  {
    long threads = (long)n_edges * 32;
    int blocks = (int)((threads + BLK - 1) / BLK);
    agnn_aggregate<<<blocks, BLK, 0, stream>>>(x, src, dst, alpha, segsum, out, n_edges);
  }
}